// MeshShuffleLIN_76819785056404
// MI455X (gfx1250) — compile-verified
//
#include <hip/hip_runtime.h>
#include <hip/hip_bf16.h>
#include <stdint.h>

#define B_      2
#define CH_     256
#define VIN_    40962
#define VOUT_   163842
#define NCHUNK_ ((VOUT_ + 31) / 32)   /* 5121 chunks of 32 vertices */

/* ---------------- CDNA5 TDM availability probe ---------------- */
#if defined(__has_builtin)
#  if __has_builtin(__builtin_amdgcn_tensor_load_to_lds) && __has_builtin(__builtin_amdgcn_s_wait_tensorcnt)
#    define USE_TDM 1
#  endif
#endif
#ifndef USE_TDM
#  define USE_TDM 0
#endif
#if USE_TDM
#  warning "CDNA5: TDM tensor_load_to_lds path ENABLED for transpose (6-arg clang-23 form)"
#else
#  warning "CDNA5: TDM builtin unavailable -> plain-load transpose fallback"
#endif

typedef unsigned int u32x4_t __attribute__((ext_vector_type(4)));
typedef int          i32x8_t __attribute__((ext_vector_type(8)));
typedef int          i32x4_t __attribute__((ext_vector_type(4)));

/* ---------------- transpose: x[b][c][v] -> xt[b][v][c] ----------------
 * One block = one 32(c) x 32(v) tile. TDM DMAs the tile into LDS with a
 * 1-dword pad per 32-dword row (stride 33), then 256 threads store the
 * transposed tile with fully coalesced 128B writes. OOB v reads return 0
 * in hardware (tensor_dim0 = remaining extent) and are never stored.    */
__global__ __launch_bounds__(256) void transpose_xk(const float* __restrict__ x,
                                                    float* __restrict__ xt) {
  __shared__ float tileT[32 * 33 + 1];   /* row stride 33 dwords */
  const int b  = blockIdx.z;
  const int v0 = blockIdx.x * 32;
  const int c0 = blockIdx.y * 32;

#if USE_TDM
  if (threadIdx.x < 32) {                /* wave 0 issues one TDM op */
    const uint64_t gaddr = (uint64_t)(uintptr_t)(x + ((size_t)(b * CH_ + c0) * VIN_ + v0));
    const uint32_t lds   = (uint32_t)(uintptr_t)(void*)&tileT[0]; /* low 32b of generic = LDS offset */
    const uint32_t td0   = (uint32_t)(VIN_ - v0);  /* remaining extent: OOB reads -> 0 */
    const uint32_t td1   = 32u;                    /* 32 channel rows, always in range */
    u32x4_t g0;
    g0.x = 1u;                                     /* count=1, user descriptor        */
    g0.y = lds;                                    /* lds_addr (bytes)                */
    g0.z = (uint32_t)gaddr;                        /* global_addr[31:0]               */
    g0.w = ((uint32_t)(gaddr >> 32) & 0x01FFFFFFu) | (2u << 30); /* addr[56:32]|type=2 */
    i32x8_t g1;
    g1[0] = (int)((2u << 16) | (1u << 20) | (4u << 22)); /* data_size=4B, pad_en, 1dw pad / 32dw */
    g1[1] = (int)((td0 & 0xFFFFu) << 16);                /* tensor_dim0[15:0]   @ bits 63:48 */
    g1[2] = (int)((td0 >> 16) | ((td1 & 0xFFFFu) << 16));/* td0[31:16] | td1[15:0]           */
    g1[3] = (int)((td1 >> 16) | (32u << 16));            /* td1[31:16] | tile_dim0=32        */
    g1[4] = 32;                                          /* tile_dim1=32, tile_dim2=0 (2D)   */
    g1[5] = VIN_;                                        /* tensor_dim0_stride[31:0]         */
    g1[6] = 0;
    g1[7] = 0;
    i32x4_t gz4 = {0, 0, 0, 0};
    i32x8_t gz8 = {0, 0, 0, 0, 0, 0, 0, 0};
    /* clang-23 / therock-10.0 6-arg form:
       (u32x4 g0, i32x8 g1, i32x4 g2, i32x4 g3, i32x8 extra, i32 cpol) */
    __builtin_amdgcn_tensor_load_to_lds(g0, g1, gz4, gz4, gz8, 0);
    __builtin_amdgcn_s_wait_tensorcnt(0);
  }
#else
  {
    const int vl = threadIdx.x & 31;
    const int cb = threadIdx.x >> 5;
#pragma unroll
    for (int k = 0; k < 4; ++k) {
      const int c = cb + k * 8;
      tileT[c * 33 + vl] =
          (v0 + vl < VIN_) ? x[(size_t)(b * CH_ + c0 + c) * VIN_ + v0 + vl] : 0.0f;
    }
  }
#endif
  __syncthreads();
  const int cc = threadIdx.x & 31;
#pragma unroll
  for (int j = 0; j < 4; ++j) {
    const int r = (threadIdx.x >> 5) + j * 8;      /* v within tile */
    if (v0 + r < VIN_)
      xt[((size_t)b * VIN_ + v0 + r) * CH_ + c0 + cc] = tileT[cc * 33 + r];
  }
}

/* ---------------- per-(b,c,v) attention kernel (pure per-lane VALU) ---- */
__device__ __forceinline__ float attn_one(const float in[8], const float* __restrict__ cf) {
  float lg[8];
#pragma unroll
  for (int f = 0; f < 8; ++f) {
    float s = in[0] * cf[f];
#pragma unroll
    for (int k = 1; k < 8; ++k) s = fmaf(in[k], cf[k * 8 + f], s);
    lg[f] = s * 0.125f;
  }
  float m = lg[0];
#pragma unroll
  for (int f = 1; f < 8; ++f) m = fmaxf(m, lg[f]);
  float sum = 0.0f, acc = 0.0f;
#pragma unroll
  for (int f = 0; f < 8; ++f) {
    const float p = __expf(lg[f] - m);
    sum += p;
    acc = fmaf(in[f], p, acc);   /* out = dot(in, softmax) ; in[k], k = e*2+d */
  }
  return acc / sum;
}

/* ---------------- main gather kernel: wave-per-vertex ----------------
 * Block = 1024 threads = 32 waves = 32 vertices per chunk. Lane L loads
 * channels [8L,8L+8) of both gathered rows (coalesced b128 loads) and
 * owns channel-groups 2L and 2L+1. Results staged in LDS, then stored
 * with 128B-coalesced writes. Next chunk's rows are prefetched.        */
__global__ __launch_bounds__(1024) void mesh_gather(const float* __restrict__ xt,
                                                    const float* __restrict__ coeffs,
                                                    const int* __restrict__ vmap,
                                                    float* __restrict__ out) {
  __shared__ float so[32][129];          /* [v_local][b*64+g], stride 129 banks-clean */
  const int lane = threadIdx.x & 31;
  const int w    = threadIdx.x >> 5;

  float cf[64];                          /* (4,2,8) flat: [e*2+d][f] -> uniform loads */
#pragma unroll
  for (int i = 0; i < 64; ++i) cf[i] = coeffs[i];

  const int stride = gridDim.x;
  int i0, i1;
  {
    const int v = min(blockIdx.x * 32 + w, VOUT_ - 1);
    i0 = vmap[2 * v];
    i1 = vmap[2 * v + 1];
  }
  for (int chunk = blockIdx.x; chunk < NCHUNK_; chunk += stride) {
    const int nxt = chunk + stride;
    int ni0 = 0, ni1 = 0;
    if (nxt < NCHUNK_) {                 /* software pipeline: load next indices,    */
      const int vn = min(nxt * 32 + w, VOUT_ - 1);
      ni0 = vmap[2 * vn];
      ni1 = vmap[2 * vn + 1];
#pragma unroll
      for (int b = 0; b < 2; ++b) {      /* ...and prefetch next gathered rows      */
        __builtin_prefetch(&xt[((size_t)b * VIN_ + ni0) * CH_ + 8 * lane], 0, 3);
        __builtin_prefetch(&xt[((size_t)b * VIN_ + ni1) * CH_ + 8 * lane], 0, 3);
      }
    }
#pragma unroll
    for (int b = 0; b < 2; ++b) {
      const float4* r0 = (const float4*)(xt + ((size_t)b * VIN_ + i0) * CH_ + 8 * lane);
      const float4* r1 = (const float4*)(xt + ((size_t)b * VIN_ + i1) * CH_ + 8 * lane);
      const float4 a0 = r0[0], a1 = r0[1];     /* d=0 rows: channels 8L..8L+7 */
      const float4 d0 = r1[0], d1 = r1[1];     /* d=1 rows                    */
      const float inA[8] = {a0.x, d0.x, a0.y, d0.y, a0.z, d0.z, a0.w, d0.w};
      const float inB[8] = {a1.x, d1.x, a1.y, d1.y, a1.z, d1.z, a1.w, d1.w};
      so[w][b * 64 + 2 * lane]     = attn_one(inA, cf);
      so[w][b * 64 + 2 * lane + 1] = attn_one(inB, cf);
    }
    __syncthreads();
#pragma unroll
    for (int j = 0; j < 4; ++j) {        /* coalesced store: 32 consecutive v per (b,g) */
      const int p  = (threadIdx.x >> 5) + j * 32;   /* p = b*64+g in [0,128) */
      const int vv = chunk * 32 + lane;
      if (vv < VOUT_) out[(size_t)p * VOUT_ + vv] = so[lane][p];
    }
    __syncthreads();
    i0 = ni0;
    i1 = ni1;
  }
}

/* ---------------- fallback if workspace too small: direct gather ------- */
__global__ __launch_bounds__(256) void mesh_direct(const float* __restrict__ x,
                                                   const float* __restrict__ coeffs,
                                                   const int* __restrict__ vmap,
                                                   float* __restrict__ out) {
  const size_t tid = (size_t)blockIdx.x * 256 + threadIdx.x;
  if (tid >= (size_t)128 * VOUT_) return;
  const int v = (int)(tid % VOUT_);
  const int p = (int)(tid / VOUT_);
  const int b = p >> 6, g = p & 63;
  const int i0 = vmap[2 * v], i1 = vmap[2 * v + 1];
  float in[8];
#pragma unroll
  for (int e = 0; e < 4; ++e) {
    const size_t row = (size_t)(b * CH_ + 4 * g + e) * VIN_;
    in[2 * e]     = x[row + i0];
    in[2 * e + 1] = x[row + i1];
  }
  float cf[64];
#pragma unroll
  for (int i = 0; i < 64; ++i) cf[i] = coeffs[i];
  out[(size_t)p * VOUT_ + v] = attn_one(in, cf);
}

/* ---------------- launch ---------------- */
extern "C" void kernel_launch(void* const* d_in, const int* in_sizes, int n_in,
                              void* d_out, int out_size, void* d_ws, size_t ws_size,
                              hipStream_t stream) {
  (void)in_sizes; (void)n_in; (void)out_size;
  const float* x      = (const float*)d_in[0];
  const float* coeffs = (const float*)d_in[1];
  const int*   vmap   = (const int*)d_in[2];
  float*       out    = (float*)d_out;

  const size_t need = (size_t)B_ * VIN_ * CH_ * sizeof(float);  /* ~84 MB */
  if (ws_size >= need) {
    float* xt = (float*)d_ws;
    dim3 gT((VIN_ + 31) / 32, CH_ / 32, B_);     /* 1281 x 8 x 2 tiles */
    transpose_xk<<<gT, 256, 0, stream>>>(x, xt);
    mesh_gather<<<1280, 1024, 0, stream>>>(xt, coeffs, vmap, out);
  } else {
    const size_t total = (size_t)128 * VOUT_;
    mesh_direct<<<(unsigned)((total + 255) / 256), 256, 0, stream>>>(x, coeffs, vmap, out);
  }
}